// GraphConvolution_67594195304484
// MI455X (gfx1250) — compile-verified
//
#include <hip/hip_runtime.h>

typedef __attribute__((ext_vector_type(2))) float v2f;
typedef __attribute__((ext_vector_type(4))) float v4f;
typedef __attribute__((ext_vector_type(8))) float v8f;

#define IN_F  128
#define OUT_F 64
#define WAVES_PER_BLOCK 8
#define THREADS 256
#define KSTEPS (IN_F / 4)            // 32 WMMA k-steps
#define NTILES (OUT_F / 16)          // 4 n-tiles
#define FRAG_FLOATS (KSTEPS * NTILES * 32 * 2)   // 8192 floats = 32 KB

// ---------------------------------------------------------------------------
// Kernel 1: support = input @ W  via V_WMMA_F32_16X16X4_F32 (exact fp32 math)
//
// Weights are staged into LDS *pre-shuffled into WMMA B-fragment order*:
//   lB flat dword index s = kstep*256 + ntile*64 + lane*2 + j
//   holds W[k][n] with k = kstep*4 + 2*(lane>>4) + j,  n = ntile*16 + (lane&15)
// so the inner loop's B fragment is a single conflict-free ds_load_b64 into an
// even-aligned VGPR pair (no repacking movs). The shuffle itself is done with
// CDNA5 GLOBAL_LOAD_ASYNC_TO_LDS_B32 (per-lane gather, ASYNCcnt-tracked).
//
// A 16x4 f32 fragment (ISA 7.12.2): lanes 0-15 hold M=lane, K={0,1};
// lanes 16-31 hold M=lane-16, K={2,3} -> one global_load_b64 per k-step.
// C/D 16x16: VGPR i holds M = i + 8*(lane>>4), N = lane&15.
// ---------------------------------------------------------------------------
__global__ __launch_bounds__(THREADS) void gc_gemm_wmma(
    const float* __restrict__ in, const float* __restrict__ W,
    float* __restrict__ support, int nNodes, int nTiles)
{
    __shared__ __align__(16) float lB[FRAG_FLOATS];   // 32 KB (of 320 KB/WGP)

    // ---- async gather-stage of W into fragment-ordered LDS ----
    {
        const int tid = threadIdx.x;
        for (int i = 0; i < FRAG_FLOATS / THREADS; ++i) {   // 32 iters
            const int s     = i * THREADS + tid;            // dest dword == task id
            const int j     = s & 1;
            const int lane  = (s >> 1) & 31;
            const int ntile = (s >> 6) & 3;
            const int kstep = s >> 8;
            const int k = kstep * 4 + 2 * (lane >> 4) + j;
            const int n = ntile * 16 + (lane & 15);
            const unsigned           ldsa = (unsigned)(uintptr_t)(&lB[s]);
            const unsigned long long ga   =
                (unsigned long long)(uintptr_t)(W + k * OUT_F + n);
            asm volatile("global_load_async_to_lds_b32 %0, %1, off"
                         :: "v"(ldsa), "v"(ga) : "memory");
        }
        asm volatile("s_wait_asynccnt 0" ::: "memory");
    }
    __syncthreads();

    const int wave = threadIdx.x >> 5;   // wave32
    const int lane = threadIdx.x & 31;
    const int tile = blockIdx.x * WAVES_PER_BLOCK + wave;
    if (tile >= nTiles) return;

    const int m0   = tile * 16;
    const int half = lane >> 4;
    const int l16  = lane & 15;

    int arow = m0 + l16;
    if (arow >= nNodes) arow = nNodes - 1;    // clamp (100000%16==0, never hit)
    const float* aptr  = in + (size_t)arow * IN_F + 2 * half;
    const float* bfrag = &lB[lane * 2];       // + kstep*256 + ntile*64 (imm offsets)

    v8f acc0 = {}, acc1 = {}, acc2 = {}, acc3 = {};

    for (int ks = 0; ks < KSTEPS; ++ks) {
        const v2f a = *(const v2f*)(aptr + ks * 4);              // global_load_b64
        const float* bp = bfrag + ks * 256;
        v2f b;
        b = *(const v2f*)(bp);                                   // ds_load_b64
        acc0 = __builtin_amdgcn_wmma_f32_16x16x4_f32(false, a, false, b,
                                                     (short)0, acc0, false, false);
        b = *(const v2f*)(bp + 64);
        acc1 = __builtin_amdgcn_wmma_f32_16x16x4_f32(false, a, false, b,
                                                     (short)0, acc1, false, false);
        b = *(const v2f*)(bp + 128);
        acc2 = __builtin_amdgcn_wmma_f32_16x16x4_f32(false, a, false, b,
                                                     (short)0, acc2, false, false);
        b = *(const v2f*)(bp + 192);
        acc3 = __builtin_amdgcn_wmma_f32_16x16x4_f32(false, a, false, b,
                                                     (short)0, acc3, false, false);
    }

    // write back: row = m0 + i + 8*half, col = ntile*16 + l16
    #pragma unroll
    for (int i = 0; i < 8; ++i) {
        const int r = m0 + 8 * half + i;
        if (r < nNodes) {
            float* o = support + (size_t)r * OUT_F + l16;
            o[0]  = acc0[i];
            o[16] = acc1[i];
            o[32] = acc2[i];
            o[48] = acc3[i];
        }
    }
}

// ---------------------------------------------------------------------------
// Kernel 2: out[n][f] = bias[f]   (seeds the scatter-sum accumulator)
// ---------------------------------------------------------------------------
__global__ __launch_bounds__(THREADS) void gc_bias_init(
    float* __restrict__ out, const float* __restrict__ bias, int total4)
{
    const int t = blockIdx.x * blockDim.x + threadIdx.x;
    if (t >= total4) return;
    const v4f b = *(const v4f*)(bias + (t & 15) * 4);
    *(v4f*)(out + (size_t)t * 4) = b;
}

// ---------------------------------------------------------------------------
// Kernel 3: COO SpMM scatter. One wave per edge; lane handles 2 features.
// support & out are L2-resident (25.6 MB each << 192 MB L2) -> gathers and
// f32 atomics resolve in L2; HBM only streams the edge arrays.
// ---------------------------------------------------------------------------
__global__ __launch_bounds__(THREADS) void gc_scatter(
    const float* __restrict__ support, const float* __restrict__ vals,
    const int* __restrict__ rows, const int* __restrict__ cols,
    float* __restrict__ out, int nEdges)
{
    const int wave = threadIdx.x >> 5;
    const int lane = threadIdx.x & 31;
    const int e = blockIdx.x * WAVES_PER_BLOCK + wave;
    if (e >= nEdges) return;

    const float v = vals[e];       // uniform per wave -> single L0 request
    const int   r = rows[e];
    const int   c = cols[e];

    const v2f s = *(const v2f*)(support + (size_t)c * OUT_F + 2 * lane);
    float* o = out + (size_t)r * OUT_F + 2 * lane;
    atomicAdd(o,     v * s.x);     // global_atomic_add_f32 (no return)
    atomicAdd(o + 1, v * s.y);
}

// ---------------------------------------------------------------------------
extern "C" void kernel_launch(void* const* d_in, const int* in_sizes, int n_in,
                              void* d_out, int out_size, void* d_ws, size_t ws_size,
                              hipStream_t stream)
{
    const float* input    = (const float*)d_in[0];
    const float* weights  = (const float*)d_in[1];
    const float* bias     = (const float*)d_in[2];
    const float* adj_vals = (const float*)d_in[3];
    const int*   adj_rows = (const int*)d_in[4];
    const int*   adj_cols = (const int*)d_in[5];
    float* out     = (float*)d_out;
    float* support = (float*)d_ws;       // nNodes*64 f32 = 25.6 MB scratch

    const int nNodes = in_sizes[0] / IN_F;   // 100000
    const int nEdges = in_sizes[3];          // 1600000

    // 1) support = input @ W
    const int nTiles = (nNodes + 15) / 16;                       // 6250
    const int gemmBlocks = (nTiles + WAVES_PER_BLOCK - 1) / WAVES_PER_BLOCK;
    gc_gemm_wmma<<<gemmBlocks, THREADS, 0, stream>>>(input, weights, support,
                                                     nNodes, nTiles);

    // 2) out = bias (broadcast)
    const int total4 = (nNodes * OUT_F) / 4;
    gc_bias_init<<<(total4 + THREADS - 1) / THREADS, THREADS, 0, stream>>>(
        out, bias, total4);

    // 3) out[r] += v * support[c]
    const int scatBlocks = (nEdges + WAVES_PER_BLOCK - 1) / WAVES_PER_BLOCK;
    gc_scatter<<<scatBlocks, THREADS, 0, stream>>>(support, adj_vals, adj_rows,
                                                   adj_cols, out, nEdges);
}